// SwapTestAttention_91199335563799
// MI455X (gfx1250) — compile-verified
//
#include <hip/hip_runtime.h>
#include <hip/hip_bf16.h>

// ---------------------------------------------------------------------------
// SwapTest attention for MI455X (gfx1250).
// Out[b,q,k] = (|prod_n conj(qs[b,q,n,:]) . ks[b,k,n,:]|^2 + 1) / 2
// Per-qubit overlap Re/Im are 4-dim real dot products  ->  8 GEMMs of
// [1024x4]x[4x1024] per batch  ->  V_WMMA_F32_16X16X4_F32 (K=4 exactly).
// ---------------------------------------------------------------------------

#define B_SZ   8
#define SEQ    1024
#define DMODEL 64
#define NQ     4
#define PI_2   1.57079632679489662f

typedef float v2f __attribute__((ext_vector_type(2)));
typedef float v8f __attribute__((ext_vector_type(8)));

// ---------------------------------------------------------------------------
// Phase 1: per-(b,pos,qubit) single-qubit state features.
//   Q side: QF[(b*SEQ+pos)*NQ + n][0..3] = [v0r, v0i, v1r, v1i]
//   K side: KF[(b*SEQ+pos)*NQ + n][re: 0..3 | im: 4..7]
//           re = [v0r, v0i, v1r, v1i],  im = [v0i, -v0r, v1i, -v1r]
// so that Re(ov) = qvec.re_vec and Im(ov) = qvec.im_vec.
// ---------------------------------------------------------------------------
__global__ void qk_state_features(const float* __restrict__ x,
                                  const float* __restrict__ theta,
                                  float* __restrict__ out, int is_k)
{
    int idx = blockIdx.x * blockDim.x + threadIdx.x;   // (b*SEQ + pos)*NQ + n
    if (idx >= B_SZ * SEQ * NQ) return;
    int n  = idx & (NQ - 1);
    int bp = idx >> 2;                                 // b*SEQ + pos

    float xv = tanhf(x[bp * DMODEL + n]);
    float a  = xv * PI_2;
    float c  = cosf(a);
    float s  = sinf(a);

    float phi = theta[n * 3 + 0];
    float th  = theta[n * 3 + 1];
    float om  = theta[n * 3 + 2];
    float ct  = cosf(0.5f * th);
    float st  = sinf(0.5f * th);
    float hp  = 0.5f * (phi + om);
    float hm  = 0.5f * (phi - om);
    float chp = cosf(hp), shp = sinf(hp);
    float chm = cosf(hm), shm = sinf(hm);

    // v0 = m00*c + m01*s ; v1 = m10*c + m11*s  (c,s real)
    float v0r =  c * ct * chp - s * st * chm;
    float v0i = -c * ct * shp - s * st * shm;
    float v1r =  c * st * chm + s * ct * chp;
    float v1i = -c * st * shm + s * ct * shp;

    if (!is_k) {
        float4 o = make_float4(v0r, v0i, v1r, v1i);
        *(float4*)(out + (size_t)idx * 4) = o;
    } else {
        float4 ore = make_float4(v0r,  v0i, v1r,  v1i);
        float4 oim = make_float4(v0i, -v0r, v1i, -v1r);
        *(float4*)(out + (size_t)idx * 8)     = ore;
        *(float4*)(out + (size_t)idx * 8 + 4) = oim;
    }
}

// ---------------------------------------------------------------------------
// Phase 2: tiled [Sq x Sk] computation. One wave32 per 16x16 output tile.
// A (16x4 f32) layout per ISA: lanes 0-15 hold K=0 (v0), K=1 (v1) of row M=lane;
// lanes 16-31 hold K=2,3 of row M=lane-16.  B (4x16) is the mirror with N=lane.
// C/D: VGPR j, lane<16 -> (M=j, N=lane); lane>=16 -> (M=j+8, N=lane-16).
// ---------------------------------------------------------------------------
__global__ void __launch_bounds__(256)
swaptest_attn_wmma(const float* __restrict__ QF,
                   const float* __restrict__ KF,
                   float* __restrict__ out)
{
    const int lane = threadIdx.x & 31;
    const int wave = threadIdx.x >> 5;

    const int tile = blockIdx.x * 8 + wave;        // 32768 tiles total
    const int b    = tile >> 12;                   // / (64*64)
    const int rem  = tile & 4095;
    const int q0   = (rem >> 6) << 4;
    const int k0   = (rem & 63) << 4;

    const int row  = lane & 15;
    const int koff = (lane < 16) ? 0 : 2;          // K split across half-waves

    const float* qbase = QF + ((size_t)(b * SEQ + q0 + row) * NQ) * 4 + koff;
    const float* kbase = KF + ((size_t)(b * SEQ + k0 + row) * NQ) * 8 + koff;

    v8f acc_re[NQ], acc_im[NQ];
#pragma unroll
    for (int n = 0; n < NQ; ++n) {
        v2f a   = *(const v2f*)(qbase + n * 4);
        v2f bre = *(const v2f*)(kbase + n * 8);
        v2f bim = *(const v2f*)(kbase + n * 8 + 4);
        v8f z   = {0.f, 0.f, 0.f, 0.f, 0.f, 0.f, 0.f, 0.f};
        acc_re[n] = __builtin_amdgcn_wmma_f32_16x16x4_f32(
            false, a, false, bre, (short)0, z, false, false);
        acc_im[n] = __builtin_amdgcn_wmma_f32_16x16x4_f32(
            false, a, false, bim, (short)0, z, false, false);
    }

    // Epilogue: amp = prod_n (re_n + i*im_n); out = (|amp|^2 + 1)/2.
    const int orow = q0 + ((lane >> 4) << 3);      // +8 for upper half-wave
    const int ocol = k0 + (lane & 15);
    float* obase = out + ((size_t)(b * SEQ + orow)) * SEQ + ocol;

#pragma unroll
    for (int j = 0; j < 8; ++j) {
        float re = acc_re[0][j];
        float im = acc_im[0][j];
#pragma unroll
        for (int n = 1; n < NQ; ++n) {
            float r2 = acc_re[n][j];
            float i2 = acc_im[n][j];
            float nr = re * r2 - im * i2;
            float ni = re * i2 + im * r2;
            re = nr;
            im = ni;
        }
        float fid = re * re + im * im;
        obase[(size_t)j * SEQ] = fid * 0.5f + 0.5f;
    }
}

extern "C" void kernel_launch(void* const* d_in, const int* in_sizes, int n_in,
                              void* d_out, int out_size, void* d_ws, size_t ws_size,
                              hipStream_t stream) {
    (void)in_sizes; (void)n_in; (void)out_size; (void)ws_size;

    const float* query   = (const float*)d_in[0];
    const float* key     = (const float*)d_in[1];
    const float* theta_q = (const float*)d_in[2];
    const float* theta_k = (const float*)d_in[3];
    float* out = (float*)d_out;

    float* QF = (float*)d_ws;                              // 8*1024*4*4 floats (0.5 MB)
    float* KF = QF + (size_t)B_SZ * SEQ * NQ * 4;          // 8*1024*4*8 floats (1.0 MB)

    const int items = B_SZ * SEQ * NQ;                     // 32768 per side
    qk_state_features<<<(items + 255) / 256, 256, 0, stream>>>(query, theta_q, QF, 0);
    qk_state_features<<<(items + 255) / 256, 256, 0, stream>>>(key,   theta_k, KF, 1);

    // 8 batches * 64*64 tiles = 32768 waves; 8 waves per 256-thread block.
    swaptest_attn_wmma<<<32768 / 8, 256, 0, stream>>>(QF, KF, out);
}